// ModelStconv_36953898615266
// MI455X (gfx1250) — compile-verified
//
#include <hip/hip_runtime.h>
#include <hip/hip_bf16.h>

typedef __attribute__((ext_vector_type(2))) float v2f;
typedef __attribute__((ext_vector_type(8))) float v8f;

#define NN 18
#define TSTEPS 1280
#define C64 64
#define T1LEN 1278              // T-2
#define T2LEN 1276              // T-4
#define M1 (T1LEN * NN)         // 23004 rows after tconv1
#define M2 (T2LEN * NN)         // 22968 rows after tconv2
#define FC_IN (M2 * C64)        // 1469952
#define MT1 ((M1 + 15) / 16)    // 1438
#define MT2 ((M2 + 15) / 16)    // 1436

// workspace layout (floats), each activation buffer padded to 1280*18 rows
#define PADROWS (TSTEPS * NN)   // 23040
#define OFF_T0 0
#define OFF_T1 (PADROWS * C64)
#define OFF_T2 (2 * PADROWS * C64)
#define OFF_SCALE (3 * PADROWS * C64)
#define OFF_SHIFT (OFF_SCALE + 32)
#define OFF_PART (OFF_SHIFT + 32)
#define NPART 1024

// ---------------- Kernel 0: zero the pad tails of T0 and T1 ------------------------
// Rows [M1, PADROWS) of T0 and T1 must be zero so the WMMA kernels can load
// unconditionally (no per-lane exec-mask churn around A-fragment loads).
__global__ void k_zero_pads(float* __restrict__ T0, float* __restrict__ T1) {
    const int padElems = (PADROWS - M1) * C64;   // 36*64 = 2304
    for (int i = threadIdx.x; i < padElems; i += blockDim.x) {
        T0[M1 * C64 + i] = 0.0f;
        T1[M1 * C64 + i] = 0.0f;
    }
}

// ---------------- Kernel 1: normalize + faithful reshape + GLU temporal conv #1 ----
// out T0[(t*18+n)*64 + c], t in [0,1278)
__global__ void k_tconv1(const float* __restrict__ x,
                         const float* __restrict__ w1a, const float* __restrict__ b1a,
                         const float* __restrict__ w1b, const float* __restrict__ b1b,
                         const float* __restrict__ w1c, const float* __restrict__ b1c,
                         float* __restrict__ T0) {
    int gid = blockIdx.x * blockDim.x + threadIdx.x;
    if (gid >= M1 * C64) return;
    int c = gid & 63;
    int m = gid >> 6;               // m = t*18 + n
    float xv[3];
#pragma unroll
    for (int k = 0; k < 3; ++k) {
        // X[t+k, n] = xn_flat[(t+k)*18 + n] = xn_flat[m + 18k]
        float xr = x[m + k * NN];
        xv[k] = xr / fmaxf(fabsf(xr), 1e-12f);
    }
    float P = b1a[c], Q = b1b[c], Cc = b1c[c];
#pragma unroll
    for (int k = 0; k < 3; ++k) {
        P  += xv[k] * w1a[c * 3 + k];
        Q  += xv[k] * w1b[c * 3 + k];
        Cc += xv[k] * w1c[c * 3 + k];
    }
    float s = 1.0f / (1.0f + expf(-Q));
    float h = P * s + Cc;
    T0[gid] = h > 0.0f ? h : 0.0f;
}

// ---------------- Kernel 2: ChebConv K=1 == GEMM (M1 x 64)x(64 x 64) + bias + ReLU -
// one wave32 per 16x16 tile, fp32 WMMA 16x16x4, K-loop of 16 steps.
// A is zero-padded past M1 rows -> unconditional b64 fragment loads.
__global__ void k_cheb_wmma(const float* __restrict__ A,   // T0 (padded)
                            const float* __restrict__ W,   // cheb_w (64x64, K-major)
                            const float* __restrict__ bias,
                            float* __restrict__ D) {       // T1 (padded)
    const int lane = threadIdx.x & 31;
    const int wave = threadIdx.x >> 5;
    const int gw   = blockIdx.x * 8 + wave;
    if (gw >= MT1 * 4) return;                 // wave-uniform
    const int mt = gw >> 2, nt = gw & 3;
    const int row0 = mt * 16;
    const int lrow = lane & 15;
    const int khi  = (lane >> 4) << 1;         // 0 or 2
    const int rowA = row0 + lrow;              // < PADROWS, zero-padded
    const int colB = nt * 16 + lrow;
    const float* __restrict__ Arow = A + rowA * 64 + khi;
    v8f acc = {};
#pragma unroll
    for (int kb = 0; kb < 64; kb += 4) {
        v2f a = *(const v2f*)(Arow + kb);      // 8B-aligned pair -> global_load_b64
        v2f b;
        b.x = W[(kb + khi) * 64 + colB];
        b.y = W[(kb + khi + 1) * 64 + colB];
        acc = __builtin_amdgcn_wmma_f32_16x16x4_f32(false, a, false, b,
                                                    (short)0, acc, false, false);
    }
    const int mhi = (lane >> 4) * 8;
    const float bb = bias[colB];
#pragma unroll
    for (int r = 0; r < 8; ++r) {
        int row = row0 + r + mhi;              // < 23008 <= PADROWS
        float v = acc[r] + bb;
        D[row * 64 + colB] = v > 0.0f ? v : 0.0f;
    }
}

// ---------------- Kernel 3: temporal conv #2 as 3x implicit-im2col GEMM + GLU ------
// A row m=(t*18+n): K=192 features = T1[m + 18*kk][ci].  B[kk*64+ci][co]=w[co,ci,0,kk]
// T1 pad rows are deterministic (relu(bias) / zeros); tiles touching them are
// stored into the T2 pad region and never read downstream.
__global__ void k_tconv2_wmma(const float* __restrict__ T1in,
                              const float* __restrict__ wa, const float* __restrict__ ba,
                              const float* __restrict__ wb, const float* __restrict__ bqb,
                              const float* __restrict__ wc, const float* __restrict__ bc,
                              float* __restrict__ T2out) {
    const int lane = threadIdx.x & 31;
    const int wave = threadIdx.x >> 5;
    const int gw   = blockIdx.x * 8 + wave;
    if (gw >= MT2 * 4) return;                 // wave-uniform
    const int mt = gw >> 2, nt = gw & 3;
    const int row0 = mt * 16;
    const int lrow = lane & 15;
    const int khi  = (lane >> 4) << 1;
    const int rowA = row0 + lrow;              // reads rows < 23012 <= PADROWS
    const int co   = nt * 16 + lrow;
    v8f accP = {}, accQ = {}, accC = {};
#pragma unroll
    for (int kk = 0; kk < 3; ++kk) {
        const float* __restrict__ Arow = T1in + (rowA + kk * NN) * 64 + khi;
        const float* __restrict__ wA = wa + co * 192 + kk;
        const float* __restrict__ wB = wb + co * 192 + kk;
        const float* __restrict__ wC = wc + co * 192 + kk;
#pragma unroll 4
        for (int kb = 0; kb < 64; kb += 4) {
            const int ci0 = kb + khi;
            v2f a = *(const v2f*)(Arow + kb);  // global_load_b64
            v2f fA, fB, fC;
            fA.x = wA[ci0 * 3];  fA.y = wA[(ci0 + 1) * 3];
            fB.x = wB[ci0 * 3];  fB.y = wB[(ci0 + 1) * 3];
            fC.x = wC[ci0 * 3];  fC.y = wC[(ci0 + 1) * 3];
            accP = __builtin_amdgcn_wmma_f32_16x16x4_f32(false, a, false, fA,
                                                         (short)0, accP, false, false);
            accQ = __builtin_amdgcn_wmma_f32_16x16x4_f32(false, a, false, fB,
                                                         (short)0, accQ, false, false);
            accC = __builtin_amdgcn_wmma_f32_16x16x4_f32(false, a, false, fC,
                                                         (short)0, accC, false, false);
        }
    }
    const int mhi = (lane >> 4) * 8;
    const float biasP = ba[co], biasQ = bqb[co], biasC = bc[co];
#pragma unroll
    for (int r = 0; r < 8; ++r) {
        int row = row0 + r + mhi;              // < 22976 <= PADROWS
        float P = accP[r] + biasP;
        float Q = accQ[r] + biasQ;
        float Cv = accC[r] + biasC;
        float s = 1.0f / (1.0f + expf(-Q));
        float h = P * s + Cv;
        T2out[row * 64 + co] = h > 0.0f ? h : 0.0f;
    }
}

// ---------------- Kernel 4: BatchNorm2d(train) per-node stats -> scale/shift -------
__global__ void k_bnstats(const float* __restrict__ T2,
                          const float* __restrict__ gamma, const float* __restrict__ beta,
                          float* __restrict__ scale, float* __restrict__ shift) {
    __shared__ float s_sum[256], s_sq[256];
    const int n = blockIdx.x;
    float sum = 0.0f, sq = 0.0f;
    for (int idx = threadIdx.x; idx < T2LEN * C64; idx += 256) {
        int t = idx >> 6, c = idx & 63;
        float v = T2[(t * NN + n) * 64 + c];
        sum += v; sq += v * v;
    }
    s_sum[threadIdx.x] = sum; s_sq[threadIdx.x] = sq;
    __syncthreads();
    for (int st = 128; st > 0; st >>= 1) {
        if (threadIdx.x < st) {
            s_sum[threadIdx.x] += s_sum[threadIdx.x + st];
            s_sq[threadIdx.x]  += s_sq[threadIdx.x + st];
        }
        __syncthreads();
    }
    if (threadIdx.x == 0) {
        const float cnt = (float)(T2LEN * C64);
        float mean = s_sum[0] / cnt;
        float var  = s_sq[0] / cnt - mean * mean;
        var = var < 0.0f ? 0.0f : var;
        float sc = gamma[n] / sqrtf(var + 1e-5f);
        scale[n] = sc;
        shift[n] = beta[n] - mean * sc;
    }
}

// ---------------- Kernel 5: fused BN-apply + ReLU + FC dot, per-block partials -----
__global__ void k_fc_partial(const float* __restrict__ T2,
                             const float* __restrict__ scale, const float* __restrict__ shift,
                             const float* __restrict__ fcw, float* __restrict__ part) {
    __shared__ float s[256];
    float acc = 0.0f;
    const int stride = gridDim.x * blockDim.x;
    for (int i = blockIdx.x * blockDim.x + threadIdx.x; i < FC_IN; i += stride) {
        int n = (i >> 6) % NN;                 // flatten order == T2 layout (t,n,c)
        float v = T2[i] * scale[n] + shift[n];
        v = v > 0.0f ? v : 0.0f;
        acc += v * fcw[i];
    }
    s[threadIdx.x] = acc; __syncthreads();
    for (int st = 128; st > 0; st >>= 1) {
        if (threadIdx.x < st) s[threadIdx.x] += s[threadIdx.x + st];
        __syncthreads();
    }
    if (threadIdx.x == 0) part[blockIdx.x] = s[0];
}

// ---------------- Kernel 6: deterministic final reduce + bias ----------------------
__global__ void k_final(const float* __restrict__ part, const float* __restrict__ fcb,
                        float* __restrict__ out) {
    __shared__ float s[256];
    float acc = 0.0f;
    for (int i = threadIdx.x; i < NPART; i += 256) acc += part[i];
    s[threadIdx.x] = acc; __syncthreads();
    for (int st = 128; st > 0; st >>= 1) {
        if (threadIdx.x < st) s[threadIdx.x] += s[threadIdx.x + st];
        __syncthreads();
    }
    if (threadIdx.x == 0) out[0] = s[0] + fcb[0];
}

extern "C" void kernel_launch(void* const* d_in, const int* in_sizes, int n_in,
                              void* d_out, int out_size, void* d_ws, size_t ws_size,
                              hipStream_t stream) {
    const float* x   = (const float*)d_in[0];
    // d_in[1] edge_index (int64), d_in[2] edge_weight: unused (ChebConv K=1 -> identity term only)
    const float* w1a = (const float*)d_in[3];
    const float* b1a = (const float*)d_in[4];
    const float* w1b = (const float*)d_in[5];
    const float* b1b = (const float*)d_in[6];
    const float* w1c = (const float*)d_in[7];
    const float* b1c = (const float*)d_in[8];
    const float* chw = (const float*)d_in[9];
    const float* chb = (const float*)d_in[10];
    const float* w2a = (const float*)d_in[11];
    const float* b2a = (const float*)d_in[12];
    const float* w2b = (const float*)d_in[13];
    const float* b2b = (const float*)d_in[14];
    const float* w2c = (const float*)d_in[15];
    const float* b2c = (const float*)d_in[16];
    const float* gma = (const float*)d_in[17];
    const float* bta = (const float*)d_in[18];
    const float* fcw = (const float*)d_in[19];
    const float* fcb = (const float*)d_in[20];

    float* out   = (float*)d_out;
    float* ws    = (float*)d_ws;
    float* T0    = ws + OFF_T0;
    float* T1    = ws + OFF_T1;
    float* T2    = ws + OFF_T2;
    float* scale = ws + OFF_SCALE;
    float* shift = ws + OFF_SHIFT;
    float* part  = ws + OFF_PART;

    k_zero_pads<<<1, 256, 0, stream>>>(T0, T1);
    k_tconv1<<<(M1 * C64 + 255) / 256, 256, 0, stream>>>(x, w1a, b1a, w1b, b1b, w1c, b1c, T0);
    k_cheb_wmma<<<(MT1 * 4 + 7) / 8, 256, 0, stream>>>(T0, chw, chb, T1);
    k_tconv2_wmma<<<(MT2 * 4 + 7) / 8, 256, 0, stream>>>(T1, w2a, b2a, w2b, b2b, w2c, b2c, T2);
    k_bnstats<<<NN, 256, 0, stream>>>(T2, gma, bta, scale, shift);
    k_fc_partial<<<NPART, 256, 0, stream>>>(T2, scale, shift, fcw, part);
    k_final<<<1, 256, 0, stream>>>(part, fcb, out);
}